// Loss_58110907515691
// MI455X (gfx1250) — compile-verified
//
#include <hip/hip_runtime.h>
#include <stdint.h>

// Problem constants (from reference): C = 81 classes. N derived from in_sizes.
#define NCLASS 81
#define ROWS_PER_TILE 32                        // 32*81*4 = 10368 B tile, 16B aligned
#define TILE_FLOATS (ROWS_PER_TILE * NCLASS)    // 2592 floats = 648 B128 chunks
#define TILE_CHUNKS (TILE_FLOATS / 4)           // 648 (<= 3*BLOCK)
#define BLOCK 256                               // 8 wave32
#define NACC (2 * NCLASS + 2)                   // num[81], cnt[81], loc_sum, pos_cnt

// ---- CDNA5 async global->LDS copy primitives (ASYNCcnt-tracked) ----
// the immediate offset is added to both the LDS and the global address -> one base
// address pair serves several chunks via the immediate offset. NT hint: data
// is streamed exactly once; don't retain in cache hierarchy.
template <int OFF>
__device__ __forceinline__ void async_g2lds_b128(uint32_t lds_off, uint64_t gaddr) {
    asm volatile("global_load_async_to_lds_b128 %0, %1, off offset:%2 th:TH_LOAD_NT"
                 :: "v"(lds_off), "v"(gaddr), "n"(OFF) : "memory");
}
__device__ __forceinline__ void async_g2lds_b32(uint32_t lds_off, uint64_t gaddr) {
    asm volatile("global_load_async_to_lds_b32 %0, %1, off th:TH_LOAD_NT"
                 :: "v"(lds_off), "v"(gaddr) : "memory");
}
__device__ __forceinline__ void wait_async0() {
    asm volatile("s_wait_asynccnt 0" ::: "memory");
}

__global__ void zero_ws_kernel(float* ws) {
    int t = threadIdx.x;
    if (t < NACC) ws[t] = 0.0f;
}

__global__ __launch_bounds__(BLOCK)
void focal_main_kernel(const float* __restrict__ conf,
                       const float* __restrict__ loc,
                       const float* __restrict__ targets,
                       const float* __restrict__ alpha,
                       float* __restrict__ ws,
                       int N, int ntiles)
{
    __shared__ float sAcc[NACC];
    __shared__ __align__(16) float sBuf[2][TILE_FLOATS];

    const int tid = threadIdx.x;
    for (int i = tid; i < NACC; i += BLOCK) sAcc[i] = 0.0f;
    __syncthreads();

    // Issue async copy of one tile (tiles are contiguous in row-major [N, C]).
    // <=3 B128 issues per thread with immediate offsets; B32 tail for generality.
    auto copy_tile = [&](int tile, float* dst) {
        int row0   = tile * ROWS_PER_TILE;
        int rows   = min(ROWS_PER_TILE, N - row0);
        int total  = rows * NCLASS;
        int chunks = total >> 2;
        const float* src = conf + (size_t)row0 * NCLASS;  // 16B aligned (32*324%16==0)
        uint32_t lds = (uint32_t)(uintptr_t)(dst + 4 * tid);
        uint64_t g   = (uint64_t)(uintptr_t)(src + 4 * tid);
        if (tid           < chunks) async_g2lds_b128<0>(lds, g);
        if (tid + BLOCK   < chunks) async_g2lds_b128<16 * BLOCK>(lds, g);
        if (tid + 2*BLOCK < chunks) async_g2lds_b128<32 * BLOCK>(lds, g);
        int rem = total & 3;
        if (tid < rem) {
            int b = chunks << 2;
            async_g2lds_b32((uint32_t)(uintptr_t)(dst + b + tid),
                            (uint64_t)(uintptr_t)(src + b + tid));
        }
    };

    int tile = blockIdx.x;                 // gridDim.x <= ntiles guaranteed by host
    copy_tile(tile, sBuf[0]);              // prologue
    wait_async0();
    __syncthreads();

    const int wave = tid >> 5;
    const int lane = tid & 31;
    int bufIdx = 0;

    for (; tile < ntiles; tile += gridDim.x) {
        int nxt = tile + gridDim.x;
        if (nxt < ntiles) copy_tile(nxt, sBuf[bufIdx ^ 1]);   // overlap copy(i+1)

        int row0 = tile * ROWS_PER_TILE;
        int rows = min(ROWS_PER_TILE, N - row0);
        const float* lbuf = sBuf[bufIdx];

        for (int r = wave; r < rows; r += BLOCK / 32) {       // one wave32 per row
            const float* rowp = lbuf + r * NCLASS;
            int gRow = row0 + r;

            // lanes cover columns l, l+32, l+64 (81 cols -> third only for l<17)
            float x0 = rowp[lane];
            float x1 = rowp[lane + 32];
            bool  h2 = (lane + 64) < NCLASS;
            float x2 = h2 ? rowp[lane + 64] : -3.0e38f;

            float mx = fmaxf(x0, fmaxf(x1, x2));
            #pragma unroll
            for (int o = 16; o > 0; o >>= 1) mx = fmaxf(mx, __shfl_xor(mx, o));

            float se = __expf(x0 - mx) + __expf(x1 - mx) + (h2 ? __expf(x2 - mx) : 0.0f);
            #pragma unroll
            for (int o = 16; o > 0; o >>= 1) se += __shfl_xor(se, o);

            int lab = 0;
            if (lane == 0)
                lab = (int)__builtin_nontemporal_load(&targets[(size_t)gRow * 5 + 4]);
            lab = __shfl(lab, 0);

            if (lane == 0 && lab >= 0) {        // focal term: only label column matters
                float xl    = rowp[lab];
                float logpt = (xl - mx) - __logf(se);
                float pt    = __expf(logpt);
                float a     = alpha[lab];
                float om    = 1.0f - pt;
                atomicAdd(&sAcc[lab], -(om * om * a * logpt));
                atomicAdd(&sAcc[NCLASS + lab], a);
            }
            if (lab > 0) {                      // smooth-L1, positive anchors only (~5%)
                if (lane < 4) {
                    float lp = __builtin_nontemporal_load(&loc[(size_t)gRow * 4 + lane]);
                    float tb = __builtin_nontemporal_load(&targets[(size_t)gRow * 5 + lane]);
                    float d  = fabsf(lp - tb);
                    float s1 = (d < 1.0f) ? 0.5f * d * d : d - 0.5f;
                    atomicAdd(&sAcc[2 * NCLASS], s1);
                }
                if (lane == 0) atomicAdd(&sAcc[2 * NCLASS + 1], 1.0f);
            }
        }

        wait_async0();        // copy(i+1) landed (per-wave); barrier publishes to WG
        __syncthreads();
        bufIdx ^= 1;
    }

    // Flush block-local accumulators to global workspace.
    for (int i = tid; i < NACC; i += BLOCK)
        if (sAcc[i] != 0.0f) atomicAdd(&ws[i], sAcc[i]);
}

__global__ void finalize_kernel(const float* __restrict__ ws, float* __restrict__ out) {
    __shared__ float red[128];
    int t = threadIdx.x;                    // 128 threads
    float v = 0.0f;
    if (t < NCLASS) v = ws[t] / fmaxf(ws[NCLASS + t], 1.0f);   // clip(denom,1,None)
    red[t] = v;
    __syncthreads();
    #pragma unroll
    for (int s = 64; s > 0; s >>= 1) {
        if (t < s) red[t] += red[t + s];
        __syncthreads();
    }
    if (t == 0) {
        out[1] = red[0];                    // conf_loss
        float ls = ws[2 * NCLASS];
        float np = ws[2 * NCLASS + 1];
        out[0] = (np > 0.0f) ? ls / fmaxf(4.0f * np, 1.0f) : 0.0f;  // loc_loss
    }
}

extern "C" void kernel_launch(void* const* d_in, const int* in_sizes, int n_in,
                              void* d_out, int out_size, void* d_ws, size_t ws_size,
                              hipStream_t stream) {
    const float* loc     = (const float*)d_in[0];  // [B,A,4]
    const float* conf    = (const float*)d_in[1];  // [B,A,C]
    const float* targets = (const float*)d_in[2];  // [B,A,5]
    const float* alpha   = (const float*)d_in[3];  // [C]
    float* ws  = (float*)d_ws;
    float* out = (float*)d_out;                    // out[0]=loc_loss, out[1]=conf_loss

    int N = in_sizes[0] / 4;                       // B*A
    int ntiles = (N + ROWS_PER_TILE - 1) / ROWS_PER_TILE;
    int blocks = ntiles < 4096 ? ntiles : 4096;

    zero_ws_kernel<<<1, 256, 0, stream>>>(ws);
    focal_main_kernel<<<blocks, BLOCK, 0, stream>>>(conf, loc, targets, alpha, ws, N, ntiles);
    finalize_kernel<<<1, 128, 0, stream>>>(ws, out);
}